// LinkPredictorGlobalModel_5102421147841
// MI455X (gfx1250) — compile-verified
//
#include <hip/hip_runtime.h>
#include <hip/hip_bf16.h>

typedef __attribute__((ext_vector_type(2))) float v2f;
typedef __attribute__((ext_vector_type(8))) float v8f;

#define NNODES 65536
#define NEDGES 1048576
#define NFEAT  (NNODES * 128)
#define CONVN  (64 * 128 * 1024)

__device__ __forceinline__ void atomAddF(float* p, float v) {
    __hip_atomic_fetch_add(p, v, __ATOMIC_RELAXED, __HIP_MEMORY_SCOPE_AGENT);
}
__device__ __forceinline__ float frelu(float v) { return v > 0.0f ? v : 0.0f; }

// ---------------- elementwise helpers ----------------
__global__ void k_fill(float* p, float v, int n) {
    int i = blockIdx.x * blockDim.x + threadIdx.x;
    if (i < n) p[i] = v;
}

__global__ void k_deg_atomic(const int* __restrict__ col, float* __restrict__ deg) {
    int e = blockIdx.x * blockDim.x + threadIdx.x;
    if (e < NEDGES) atomAddF(&deg[col[e]], 1.0f);
}

__global__ void k_rsqrt_inplace(float* p, int n) {
    int i = blockIdx.x * blockDim.x + threadIdx.x;
    if (i < n) p[i] = rsqrtf(p[i]);
}

// agg[i][f] = dinv[i]^2 * hw[i][f]   (self-loop contribution, also zero-inits agg)
__global__ void k_selfloop(const float* __restrict__ hw, const float* __restrict__ dinv,
                           float* __restrict__ agg) {
    int i = blockIdx.x * blockDim.x + threadIdx.x;
    if (i < NFEAT) {
        float d = dinv[i >> 7];
        agg[i] = d * d * hw[i];
    }
}

// one wave per edge: agg[col] += dinv[row]*dinv[col] * hw[row]  (128 feats, 4/lane)
__global__ void __launch_bounds__(256) k_edge_agg(const int* __restrict__ ei,
                                                  const float* __restrict__ dinv,
                                                  const float* __restrict__ hw,
                                                  float* __restrict__ agg) {
    int wid  = threadIdx.x >> 5;
    int lane = threadIdx.x & 31;
    int e = blockIdx.x * 8 + wid;
    if (e >= NEDGES) return;
    int r = ei[e];
    int c = ei[NEDGES + e];
    float nrm = dinv[r] * dinv[c];
    const float4* src = (const float4*)(hw + (size_t)r * 128);
    float4 v = src[lane];
    float* dst = agg + (size_t)c * 128 + lane * 4;
    atomAddF(dst + 0, nrm * v.x);
    atomAddF(dst + 1, nrm * v.y);
    atomAddF(dst + 2, nrm * v.z);
    atomAddF(dst + 3, nrm * v.w);
}

__global__ void k_bias_relu(float* __restrict__ p, const float* __restrict__ b) {
    int i = blockIdx.x * blockDim.x + threadIdx.x;
    if (i < NFEAT) p[i] = frelu(p[i] + b[i & 127]);
}

// ---------------- fp32 WMMA GEMM: out[M,128] = A[M,128-block] @ B[K,128] ----------------
// Block computes a 128x128 tile (8 waves x 16 rows); A and B staged in LDS by K-panels.
template <int K>
__global__ void __launch_bounds__(256) k_gemm_wmma(const float* __restrict__ A,
                                                   const float* __restrict__ B,
                                                   float* __restrict__ out) {
    constexpr int KP = (K < 32) ? K : 32;   // K-panel width
    constexpr int AS = KP + 1;              // padded A row stride (kills bank conflicts)
    __shared__ float Al[128 * AS];
    __shared__ float Bls[KP * 128];

    int tid  = threadIdx.x;
    int wid  = tid >> 5;
    int lane = tid & 31;
    int lm = lane & 15;
    int kh = (lane < 16) ? 0 : 2;
    int mBlock = blockIdx.x * 128;
    int m0 = wid * 16;                      // local row base

    v8f acc[8];
    for (int nt = 0; nt < 8; ++nt)
        for (int r = 0; r < 8; ++r) acc[nt][r] = 0.0f;

    for (int kp = 0; kp < K; kp += KP) {
        for (int i = tid; i < 128 * KP; i += 256) {
            int row = i / KP, kk = i - row * KP;
            Al[row * AS + kk] = A[(size_t)(mBlock + row) * K + kp + kk];
        }
        for (int i = tid; i < KP * 128; i += 256)
            Bls[i] = B[(size_t)kp * 128 + i];
        __syncthreads();

        for (int k0 = 0; k0 < KP; k0 += 4) {
            v2f a;
            a.x = Al[(m0 + lm) * AS + k0 + kh];
            a.y = Al[(m0 + lm) * AS + k0 + kh + 1];
            for (int nt = 0; nt < 8; ++nt) {
                int n = nt * 16 + lm;
                v2f bf;
                bf.x = Bls[(k0 + kh) * 128 + n];
                bf.y = Bls[(k0 + kh + 1) * 128 + n];
                acc[nt] = __builtin_amdgcn_wmma_f32_16x16x4_f32(
                    false, a, false, bf, (short)0, acc[nt], false, false);
            }
        }
        __syncthreads();
    }

    int ra = (lane < 16) ? 0 : 8;
    int gm = mBlock + m0;
    for (int nt = 0; nt < 8; ++nt)
        for (int r = 0; r < 8; ++r)
            out[(size_t)(gm + r + ra) * 128 + nt * 16 + lm] = acc[nt][r];
}

// ---------------- first conv (C_IN=4, k=3) + BN + relu, scalar ----------------
__global__ void k_conv_in4(const float* __restrict__ x, const float* __restrict__ w,
                           const float* __restrict__ bias, const float* __restrict__ g,
                           const float* __restrict__ be, float* __restrict__ out) {
    int i = blockIdx.x * blockDim.x + threadIdx.x;
    if (i >= CONVN) return;
    int s = i & 1023;
    int o = (i >> 10) & 127;
    int b = i >> 17;
    float acc = bias[o];
    const float* wo = w + o * 12;  // [4][3]
    for (int t = 0; t < 3; ++t) {
        int sin = s + t - 1;
        if (sin < 0 || sin >= 1024) continue;
        const float* xp = x + (size_t)(b * 1024 + sin) * 4;
        acc += wo[0 * 3 + t] * xp[0] + wo[1 * 3 + t] * xp[1] +
               wo[2 * 3 + t] * xp[2] + wo[3 * 3 + t] * xp[3];
    }
    out[i] = frelu(g[o] * acc + be[o]);
}

// ---------------- 128->128 k=3 conv as WMMA GEMM over taps ----------------
// Block = (batch, 64-position chunk); 8 waves cover the 8 out-channel tiles and
// share the input slab [128ch x 66pos] staged (and zero-padded) in LDS.
// MODE 0: out = relu(g*(acc+bias)+be + 1x1(xin))        -> store
// MODE 1: out = relu(g*(acc+bias)+be)                   -> store
// MODE 2: v = relu(g*(acc+bias)+be + resid); sum_s v -> atomicAdd out[b*128+o]
template <int MODE>
__global__ void __launch_bounds__(256) k_conv_wmma(
    const float* __restrict__ in,    // [64][128][1024]
    const float* __restrict__ w,     // [128][128][3]
    const float* __restrict__ bias, const float* __restrict__ g,
    const float* __restrict__ be,
    const float* __restrict__ xin,   // MODE 0: node-major x [65536][4]
    const float* __restrict__ scw,   // MODE 0: [128][4]
    const float* __restrict__ scb,   // MODE 0: [128]
    const float* __restrict__ resid, // MODE 2: [64][128][1024]
    float* __restrict__ out) {
    constexpr int SW = 68;                  // padded LDS row stride (66 valid cols)
    __shared__ float Sl[128 * SW];

    int tid  = threadIdx.x;
    int wid  = tid >> 5;                    // = o-tile
    int lane = tid & 31;
    int b     = blockIdx.x >> 4;
    int chunk = blockIdx.x & 15;
    int s0 = chunk * 64;
    int lm = lane & 15;
    int kh = (lane < 16) ? 0 : 2;
    int m0 = wid * 16;
    const float* inb = in + (size_t)b * 131072;

    // stage input slab: col j <-> position s0-1+j, j in [0,65], zero outside [0,1024)
    for (int i = tid; i < 128 * 66; i += 256) {
        int c = i / 66, j = i - c * 66;
        int sp = s0 - 1 + j;
        Sl[c * SW + j] = (sp >= 0 && sp < 1024) ? inb[c * 1024 + sp] : 0.0f;
    }
    __syncthreads();

    v8f acc[4];
    for (int nt = 0; nt < 4; ++nt)
        for (int r = 0; r < 8; ++r) acc[nt][r] = 0.0f;

    const float* wrow = w + (size_t)(m0 + lm) * 384;  // [c][t] for this out channel
    for (int t = 0; t < 3; ++t) {
        for (int k0 = 0; k0 < 128; k0 += 4) {
            v2f a;
            a.x = wrow[(k0 + kh) * 3 + t];
            a.y = wrow[(k0 + kh + 1) * 3 + t];
            for (int nt = 0; nt < 4; ++nt) {
                int j = nt * 16 + lm + t;   // (lm + t-1) + 1 offset into slab
                v2f bf;
                bf.x = Sl[(k0 + kh) * SW + j];
                bf.y = Sl[(k0 + kh + 1) * SW + j];
                acc[nt] = __builtin_amdgcn_wmma_f32_16x16x4_f32(
                    false, a, false, bf, (short)0, acc[nt], false, false);
            }
        }
    }

    int ra = (lane < 16) ? 0 : 8;
    float rsum[8];
    for (int r = 0; r < 8; ++r) rsum[r] = 0.0f;

    for (int nt = 0; nt < 4; ++nt) {
        int s = s0 + nt * 16 + lm;
        for (int r = 0; r < 8; ++r) {
            int o = m0 + r + ra;
            float v = g[o] * (acc[nt][r] + bias[o]) + be[o];
            if (MODE == 0) {
                const float* xp = xin + (size_t)(b * 1024 + s) * 4;
                v += scw[o * 4 + 0] * xp[0] + scw[o * 4 + 1] * xp[1] +
                     scw[o * 4 + 2] * xp[2] + scw[o * 4 + 3] * xp[3] + scb[o];
            }
            if (MODE == 2) v += resid[(size_t)b * 131072 + (size_t)o * 1024 + s];
            v = frelu(v);
            if (MODE == 2) rsum[r] += v;
            else out[(size_t)b * 131072 + (size_t)o * 1024 + s] = v;
        }
    }

    if (MODE == 2) {
        for (int r = 0; r < 8; ++r) {
            float v = rsum[r];
            v += __shfl_xor(v, 1, 32);
            v += __shfl_xor(v, 2, 32);
            v += __shfl_xor(v, 4, 32);
            v += __shfl_xor(v, 8, 32);  // full sum within each 16-lane half
            if (lm == 0) atomAddF(&out[b * 128 + m0 + r + ra], v);
        }
    }
}

__global__ void k_mean_finish(const float* __restrict__ acc, float* __restrict__ out, int n) {
    int i = blockIdx.x * blockDim.x + threadIdx.x;
    if (i < n) out[i] = acc[i] * (1.0f / 1024.0f);
}

// ---------------- launch ----------------
extern "C" void kernel_launch(void* const* d_in, const int* in_sizes, int n_in,
                              void* d_out, int out_size, void* d_ws, size_t ws_size,
                              hipStream_t stream) {
    const float* x    = (const float*)d_in[0];
    const int*   ei   = (const int*)d_in[1];   // [2][E]
    const float* gW1  = (const float*)d_in[3];
    const float* gb1  = (const float*)d_in[4];
    const float* gW2  = (const float*)d_in[5];
    const float* gb2  = (const float*)d_in[6];
    const float* r1w1 = (const float*)d_in[7];
    const float* r1b1 = (const float*)d_in[8];
    const float* r1g1 = (const float*)d_in[9];
    const float* r1be1= (const float*)d_in[10];
    const float* r1w2 = (const float*)d_in[11];
    const float* r1b2 = (const float*)d_in[12];
    const float* r1g2 = (const float*)d_in[13];
    const float* r1be2= (const float*)d_in[14];
    const float* r1sw = (const float*)d_in[15];
    const float* r1sb = (const float*)d_in[16];
    const float* r2w1 = (const float*)d_in[17];
    const float* r2b1 = (const float*)d_in[18];
    const float* r2g1 = (const float*)d_in[19];
    const float* r2be1= (const float*)d_in[20];
    const float* r2w2 = (const float*)d_in[21];
    const float* r2b2 = (const float*)d_in[22];
    const float* r2g2 = (const float*)d_in[23];
    const float* r2be2= (const float*)d_in[24];

    float* ws   = (float*)d_ws;
    float* dinv = ws;                    // 65536 f
    float* bufA = ws + 65536;            // 8388608 f
    float* bufB = bufA + NFEAT;          // 8388608 f
    float* macc = bufB + NFEAT;          // 8192 f
    float* hout = (float*)d_out;         // 8388608 f
    float* gout = hout + NFEAT;          // 8192 f

    // ---- GCN ----
    k_fill<<<NNODES / 256, 256, 0, stream>>>(dinv, 1.0f, NNODES);        // self-loop deg
    k_deg_atomic<<<NEDGES / 256, 256, 0, stream>>>(ei + NEDGES, dinv);
    k_rsqrt_inplace<<<NNODES / 256, 256, 0, stream>>>(dinv, NNODES);

    k_gemm_wmma<4><<<512, 256, 0, stream>>>(x, gW1, bufA);               // hw1
    k_selfloop<<<NFEAT / 256, 256, 0, stream>>>(bufA, dinv, bufB);
    k_edge_agg<<<NEDGES / 8, 256, 0, stream>>>(ei, dinv, bufA, bufB);
    k_bias_relu<<<NFEAT / 256, 256, 0, stream>>>(bufB, gb1);             // h1

    k_gemm_wmma<128><<<512, 256, 0, stream>>>(bufB, gW2, bufA);          // hw2
    k_selfloop<<<NFEAT / 256, 256, 0, stream>>>(bufA, dinv, hout);
    k_edge_agg<<<NEDGES / 8, 256, 0, stream>>>(ei, dinv, bufA, hout);
    k_bias_relu<<<NFEAT / 256, 256, 0, stream>>>(hout, gb2);             // h -> d_out

    // ---- conv stack ----
    k_conv_in4<<<CONVN / 256, 256, 0, stream>>>(x, r1w1, r1b1, r1g1, r1be1, bufA); // y1
    k_conv_wmma<0><<<1024, 256, 0, stream>>>(bufA, r1w2, r1b2, r1g2, r1be2,
                                             x, r1sw, r1sb, nullptr, bufB);          // f1
    k_conv_wmma<1><<<1024, 256, 0, stream>>>(bufB, r2w1, r2b1, r2g1, r2be1,
                                             nullptr, nullptr, nullptr, nullptr, bufA); // y3
    k_fill<<<32, 256, 0, stream>>>(macc, 0.0f, 8192);
    k_conv_wmma<2><<<1024, 256, 0, stream>>>(bufA, r2w2, r2b2, r2g2, r2be2,
                                             nullptr, nullptr, nullptr, bufB, macc);  // f2 -> row sums
    k_mean_finish<<<32, 256, 0, stream>>>(macc, gout, 8192);
}